// PatchLayer_73143293051537
// MI455X (gfx1250) — compile-verified
//
#include <hip/hip_runtime.h>
#include <stdint.h>

// TDM descriptor SGPR groups (amdgpu-toolchain clang-23 builtin signature:
//   tensor_load_to_lds(uint32x4 g0, int32x8 g1, int32x4 g2, int32x4 g3, int32x8 extra, i32 cpol))
typedef unsigned int v4u __attribute__((ext_vector_type(4)));
typedef int          v8i __attribute__((ext_vector_type(8)));
typedef int          v4i __attribute__((ext_vector_type(4)));

#define PATCHES_PER_WG 8          // 8 consecutive pw patches; never crosses a ph row (32 % 8 == 0)
#define PATCH_FLOATS   768        // 16*16*3
#define ROW_FLOATS     1536       // 512*3
#define PATCH_COL_FLTS 48         // 16*3

// One wave32 per workgroup. The wave only builds two Tensor-DMA descriptors:
//  1) tensor_load_to_lds : 3D tile {x=48, y=16 (stride 1536), z=8 (stride 48)} -> LDS,
//     which lands in LDS already in output (patch-major) order.
//  2) tensor_store_from_lds: one contiguous 6144-float (24 KB) store to the output.
// All data movement is done by the TDM; dependencies via TENSORcnt.
__global__ __launch_bounds__(32) void PatchLayer_tdm_kernel(const float* __restrict__ in,
                                                            float* __restrict__ out) {
    __shared__ float lds_tile[PATCHES_PER_WG * PATCH_FLOATS];   // 24 KB

    const uint32_t g0id   = blockIdx.x * PATCHES_PER_WG;  // first global patch id
    const uint32_t b      = g0id >> 10;                   // 1024 patches per image
    const uint32_t within = g0id & 1023u;
    const uint32_t ph     = within >> 5;                  // 32 patches per image row
    const uint32_t pw0    = within & 31u;                 // multiple of 8

    // Byte addresses of the tile start (input) and contiguous destination (output).
    const uint64_t in_addr = (uint64_t)(uintptr_t)in +
        (((uint64_t)b * (512ull * 512ull * 3ull)) +
         ((uint64_t)ph * 16ull * (uint64_t)ROW_FLOATS) +
         ((uint64_t)pw0 * (uint64_t)PATCH_COL_FLTS)) * 4ull;
    const uint64_t out_addr = (uint64_t)(uintptr_t)out +
        (uint64_t)g0id * (uint64_t)PATCH_FLOATS * 4ull;

    // LDS byte offset within this workgroup's allocation (low 32 bits of the
    // generic address of an addrspace(3) object are the LDS offset).
    const uint32_t lds_off = (uint32_t)(uintptr_t)&lds_tile[0];

    // Unused trailing descriptor words (beyond the 4 architectural D# groups).
    const v8i zeros8 = {0, 0, 0, 0, 0, 0, 0, 0};

    // ---------------- load descriptor: global -> LDS ----------------
    v4u ld_g0;
    ld_g0[0] = 1u;                                   // count=1 (valid), user mode, no gather
    ld_g0[1] = lds_off;                              // lds_addr
    ld_g0[2] = (uint32_t)in_addr;                    // global_addr[31:0]
    ld_g0[3] = ((uint32_t)(in_addr >> 32) & 0x01FFFFFFu) | 0x80000000u; // addr[56:32] | type=2

    v8i ld_g1;
    ld_g1[0] = (int)(2u << 16);                      // workgroup_mask=0, data_size=2 (4 bytes)
    ld_g1[1] = (int)(48u << 16);                     // tensor_dim0 = 48  (bits 79:48)
    ld_g1[2] = (int)(16u << 16);                     // tensor_dim1 = 16  (bits 111:80)
    ld_g1[3] = (int)(48u << 16);                     // tile_dim0   = 48  (bits 127:112)
    ld_g1[4] = (int)(16u | (8u << 16));              // tile_dim1=16, tile_dim2=8
    ld_g1[5] = (int)1536u;                           // tensor_dim0_stride = 1536 floats (row)
    ld_g1[6] = (int)(48u << 16);                     // tensor_dim1_stride = 48 floats (next pw patch)
    ld_g1[7] = 0;

    v4i ld_g2;
    ld_g2[0] = 8;                                    // tensor_dim2 = 8 (z extent)
    ld_g2[1] = 1;                                    // tensor_dim3 = 1 (unused, non-zero for OOB)
    ld_g2[2] = 0;                                    // tensor_dim2_stride = 0 (no 4th dim)
    ld_g2[3] = 0;                                    // tile_dim3 = 0 (unused)

    v4i ld_g3;
    ld_g3[0] = 0;                                    // tensor_dim3_stride
    ld_g3[1] = (int)(1u << 16);                      // tensor_dim4 = 1 (bits 79:48)
    ld_g3[2] = 0;                                    // tile_dim4 = 0
    ld_g3[3] = 0;

    __builtin_amdgcn_tensor_load_to_lds(ld_g0, ld_g1, ld_g2, ld_g3, zeros8, 0);

    // Wait for the tile to land in LDS before the dependent store descriptor runs.
    __builtin_amdgcn_s_wait_tensorcnt(0);

    // ---------------- store descriptor: LDS -> global (contiguous) ----------------
    v4u st_g0;
    st_g0[0] = 1u;
    st_g0[1] = lds_off;
    st_g0[2] = (uint32_t)out_addr;
    st_g0[3] = ((uint32_t)(out_addr >> 32) & 0x01FFFFFFu) | 0x80000000u;

    v8i st_g1;
    st_g1[0] = (int)(2u << 16);                      // data_size = 4 bytes
    st_g1[1] = (int)(6144u << 16);                   // tensor_dim0 = 6144 floats
    st_g1[2] = (int)(1u << 16);                      // tensor_dim1 = 1
    st_g1[3] = (int)(6144u << 16);                   // tile_dim0 = 6144 (fits 16 bits)
    st_g1[4] = 1;                                    // tile_dim1 = 1, tile_dim2 = 0
    st_g1[5] = (int)6144u;                           // tensor_dim0_stride (degenerate)
    st_g1[6] = 0;
    st_g1[7] = 0;

    v4i st_g2;
    st_g2[0] = 1;                                    // tensor_dim2 = 1 (unused, safe)
    st_g2[1] = 1;                                    // tensor_dim3 = 1
    st_g2[2] = 0;
    st_g2[3] = 0;

    v4i st_g3;
    st_g3[0] = 0;
    st_g3[1] = (int)(1u << 16);                      // tensor_dim4 = 1
    st_g3[2] = 0;
    st_g3[3] = 0;

    __builtin_amdgcn_tensor_store_from_lds(st_g0, st_g1, st_g2, st_g3, zeros8, 0);

    // Drain before LDS deallocation at wave end (S_ENDPGM also implies wait-idle).
    __builtin_amdgcn_s_wait_tensorcnt(0);
}

extern "C" void kernel_launch(void* const* d_in, const int* in_sizes, int n_in,
                              void* d_out, int out_size, void* d_ws, size_t ws_size,
                              hipStream_t stream) {
    (void)in_sizes; (void)n_in; (void)out_size; (void)d_ws; (void)ws_size;
    const float* in  = (const float*)d_in[0];
    float*       out = (float*)d_out;

    // 64 images * 1024 patches = 65536 patches, 8 per workgroup.
    const int num_blocks = (64 * 1024) / PATCHES_PER_WG;   // 8192
    PatchLayer_tdm_kernel<<<num_blocks, 32, 0, stream>>>(in, out);
}